// attHGNN_12249246728869
// MI455X (gfx1250) — compile-verified
//
#include <hip/hip_runtime.h>
#include <hip/hip_bf16.h>

// Problem constants (from reference)
#define NN 100000   // nodes
#define TT 17       // edge types
#define EE 100000   // edges per type
#define DD 128      // in feats
#define DOUT 128    // out feats

typedef float f32x2 __attribute__((ext_vector_type(2)));
typedef float f32x8 __attribute__((ext_vector_type(8)));

// ---------------------------------------------------------------------------
// Kernel 0: zero workspace (h_new[N*D] floats + cnt[T*N] ints), int4 stores
// ---------------------------------------------------------------------------
__global__ __launch_bounds__(256) void attHGNN_zero_kernel(int4* __restrict__ p, int n4) {
    int i = blockIdx.x * 256 + threadIdx.x;
    if (i < n4) p[i] = make_int4(0, 0, 0, 0);
}

// ---------------------------------------------------------------------------
// Kernel 1: per-(type, dst) edge counts via int atomics
// ---------------------------------------------------------------------------
__global__ __launch_bounds__(256) void attHGNN_count_kernel(const int* __restrict__ dst,
                                                            int* __restrict__ cnt) {
    int tid = blockIdx.x * 256 + threadIdx.x;
    if (tid < TT * EE) {
        int t = tid / EE;
        int dn = dst[tid];
        atomicAdd(&cnt[t * NN + dn], 1);
    }
}

// ---------------------------------------------------------------------------
// Kernel 2: fused scatter:
//   h_new[dst] += (tw2[t] / max(cnt[t,dst],1)) * tw[t] .* h[src]
// One wave per edge; lane handles 4 consecutive floats (float4 gather,
// 4 coalesced global_atomic_add_f32). h and h_new are L2-resident (51MB each).
// ---------------------------------------------------------------------------
__global__ __launch_bounds__(256) void attHGNN_scatter_kernel(
        const float* __restrict__ h, const int* __restrict__ src,
        const int* __restrict__ dst, const float* __restrict__ tw,
        const float* __restrict__ tw2, const int* __restrict__ cnt,
        float* __restrict__ hnew) {
    const int lane = threadIdx.x & 31;
    const int edge = blockIdx.x * 8 + (threadIdx.x >> 5);   // 8 waves per block
    if (edge >= TT * EE) return;
    const int t = edge / EE;

    const int s  = src[edge];
    const int dn = dst[edge];
    const float c = (float)cnt[t * NN + dn];
    const float scale = tw2[t] / fmaxf(c, 1.0f);

    const float4 hv  = ((const float4*)(h  + (long)s * DD))[lane];
    const float4 twv = ((const float4*)(tw + t * DD))[lane];
    float* o = hnew + (long)dn * DD + lane * 4;
    unsafeAtomicAdd(o + 0, scale * hv.x * twv.x);
    unsafeAtomicAdd(o + 1, scale * hv.y * twv.y);
    unsafeAtomicAdd(o + 2, scale * hv.z * twv.z);
    unsafeAtomicAdd(o + 3, scale * hv.w * twv.w);
}

// ---------------------------------------------------------------------------
// Kernel 3: out[N,128] = concat(h, h_new)[N,256] @ W[128,256]^T + b
// via V_WMMA_F32_16X16X4_F32. Block = 8 waves = one 16-row stripe;
// wave w computes output cols [16w, 16w+16). K=256 -> 64 chained WMMAs.
// No divergence anywhere: EXEC all-ones as WMMA requires.
// ---------------------------------------------------------------------------
__global__ __launch_bounds__(256) void attHGNN_gemm_kernel(
        const float* __restrict__ h, const float* __restrict__ hnew,
        const float* __restrict__ W, const float* __restrict__ bias,
        float* __restrict__ out) {
    const int lane  = threadIdx.x & 31;
    const int wave  = threadIdx.x >> 5;       // 0..7
    const int m0    = blockIdx.x * 16;        // row tile (6250 blocks exactly)
    const int n0    = wave * 16;              // col tile
    const int l16   = lane & 15;
    const int lhalf = lane >> 4;              // 0 or 1 -> K offset +0 / +2

    // A: lane holds row m0+l16, two consecutive K values at k0 + 2*lhalf.
    const float* Ah = h    + (long)(m0 + l16) * DD;
    const float* An = hnew + (long)(m0 + l16) * DD;
    // B = W^T: B[k][n] = W[n*256 + k]; lane holds col n0+l16.
    const float* Bw = W + (long)(n0 + l16) * (2 * DD);

    f32x8 acc = {};
    // First 128 K-values come from h, next 128 from h_new.
#pragma unroll
    for (int k0 = 0; k0 < DD; k0 += 4) {
        const int k = k0 + lhalf * 2;
        f32x2 a = *(const f32x2*)(Ah + k);
        f32x2 b = *(const f32x2*)(Bw + k);
        acc = __builtin_amdgcn_wmma_f32_16x16x4_f32(
                  false, a, false, b, (short)0, acc, false, false);
    }
#pragma unroll
    for (int k0 = 0; k0 < DD; k0 += 4) {
        const int k = k0 + lhalf * 2;
        f32x2 a = *(const f32x2*)(An + k);
        f32x2 b = *(const f32x2*)(Bw + DD + k);
        acc = __builtin_amdgcn_wmma_f32_16x16x4_f32(
                  false, a, false, b, (short)0, acc, false, false);
    }

    const float bv = bias[n0 + l16];
    // C/D layout: VGPR r, lane l -> row m0 + r + 8*lhalf, col n0 + l16
#pragma unroll
    for (int r = 0; r < 8; ++r) {
        out[(long)(m0 + r + 8 * lhalf) * DOUT + n0 + l16] = acc[r] + bv;
    }
}

// ---------------------------------------------------------------------------
extern "C" void kernel_launch(void* const* d_in, const int* in_sizes, int n_in,
                              void* d_out, int out_size, void* d_ws, size_t ws_size,
                              hipStream_t stream) {
    const float* h   = (const float*)d_in[0];
    const int*   src = (const int*)  d_in[1];
    const int*   dst = (const int*)  d_in[2];
    const float* tw  = (const float*)d_in[3];
    const float* tw2 = (const float*)d_in[4];
    const float* W   = (const float*)d_in[5];
    const float* b   = (const float*)d_in[6];
    float* out = (float*)d_out;

    // Workspace layout: [0, N*D) floats = h_new ; then [T*N) ints = cnt
    float* hnew = (float*)d_ws;
    int*   cnt  = (int*)((char*)d_ws + (size_t)NN * DD * sizeof(float));

    // Zero h_new + cnt (14.5M words, divisible by 4)
    const int n4 = (NN * DD + TT * NN) / 4;
    attHGNN_zero_kernel<<<(n4 + 255) / 256, 256, 0, stream>>>((int4*)d_ws, n4);

    // Per-(type,dst) counts
    attHGNN_count_kernel<<<(TT * EE + 255) / 256, 256, 0, stream>>>(dst, cnt);

    // Fused scale + segment-mean + type-mix scatter (1 wave per edge)
    attHGNN_scatter_kernel<<<(TT * EE) / 8, 256, 0, stream>>>(
        h, src, dst, tw, tw2, cnt, hnew);

    // WMMA GEMM: out = [h | h_new] @ W^T + b  (6250 row stripes of 16)
    attHGNN_gemm_kernel<<<NN / 16, 256, 0, stream>>>(h, hnew, W, b, out);
}